// SGC_22110491640592
// MI455X (gfx1250) — compile-verified
//
#include <hip/hip_runtime.h>

typedef float v2f __attribute__((ext_vector_type(2)));
typedef float v8f __attribute__((ext_vector_type(8)));

#define N_NODES 100000
#define N_EDGES 1600000
#define NFEAT   128
#define NCLASS  64

// ---------------------------------------------------------------------------
// Init: H1 = 0 (accumulator for hop 1), out = broadcast bias (accumulator for
// hop 2, so the bias add is free).
// ---------------------------------------------------------------------------
__global__ void sgc_init_kernel(float* __restrict__ h1,
                                float* __restrict__ out,
                                const float* __restrict__ bias) {
    int i = blockIdx.x * blockDim.x + threadIdx.x;
    if (i < N_NODES * NCLASS) {
        h1[i]  = 0.0f;
        out[i] = bias[i & (NCLASS - 1)];
    }
}

// ---------------------------------------------------------------------------
// H0 = X @ W^T  using V_WMMA_F32_16X16X4_F32 (exact f32 math).
// One wave computes a 16-node x 64-class tile: 4 accumulators (N-tiles of 16),
// 32 K-steps of 4. 256-thread block = 8 waves = 128 rows per block.
// N_NODES % 16 == 0, so every active wave has a full tile (EXEC all ones).
// ---------------------------------------------------------------------------
__global__ __launch_bounds__(256) void sgc_gemm_kernel(
        const float* __restrict__ x,
        const float* __restrict__ W,
        float* __restrict__ h0) {
    const int lane = threadIdx.x & 31;
    const int wave = threadIdx.x >> 5;
    const int m    = lane & 15;   // row within tile (A) / class within tile (B,D col)
    const int half = lane >> 4;   // K sub-offset selector

    const long base = ((long)blockIdx.x * 8 + wave) * 16;
    if (base >= N_NODES) return;  // wave-uniform exit

    v8f acc0 = {0.f,0.f,0.f,0.f,0.f,0.f,0.f,0.f};
    v8f acc1 = acc0, acc2 = acc0, acc3 = acc0;

    // A: lane holds x[base+m][k0 + 2*half + {0,1}]  (contiguous float2)
    const float* ap  = x + (size_t)(base + m) * NFEAT + 2 * half;
    // B: lane holds W[tile*16 + m][k0 + 2*half + {0,1}]
    const float* wp0 = W + (size_t)(0  + m) * NFEAT + 2 * half;
    const float* wp1 = W + (size_t)(16 + m) * NFEAT + 2 * half;
    const float* wp2 = W + (size_t)(32 + m) * NFEAT + 2 * half;
    const float* wp3 = W + (size_t)(48 + m) * NFEAT + 2 * half;

#pragma unroll 4
    for (int k0 = 0; k0 < NFEAT; k0 += 4) {
        v2f a  = *(const v2f*)(ap  + k0);
        v2f b0 = *(const v2f*)(wp0 + k0);
        v2f b1 = *(const v2f*)(wp1 + k0);
        v2f b2 = *(const v2f*)(wp2 + k0);
        v2f b3 = *(const v2f*)(wp3 + k0);
        acc0 = __builtin_amdgcn_wmma_f32_16x16x4_f32(false, a, false, b0,
                                                     (short)0, acc0, false, false);
        acc1 = __builtin_amdgcn_wmma_f32_16x16x4_f32(false, a, false, b1,
                                                     (short)0, acc1, false, false);
        acc2 = __builtin_amdgcn_wmma_f32_16x16x4_f32(false, a, false, b2,
                                                     (short)0, acc2, false, false);
        acc3 = __builtin_amdgcn_wmma_f32_16x16x4_f32(false, a, false, b3,
                                                     (short)0, acc3, false, false);
    }

    // D layout: VGPR r <-> row (base + r + 8*half), col (lane & 15)
#pragma unroll
    for (int r = 0; r < 8; ++r) {
        const size_t row = (size_t)base + r + 8 * half;
        float* o = h0 + row * NCLASS + m;
        o[0]  = acc0[r];
        o[16] = acc1[r];
        o[32] = acc2[r];
        o[48] = acc3[r];
    }
}

// ---------------------------------------------------------------------------
// One hop of COO SpMM with 64-wide features: dst[row] += val * src[col].
// One wave per edge; each lane handles a float2 (coalesced 256B gather),
// scatter via native f32 global atomics (no CAS loop).
// ---------------------------------------------------------------------------
__global__ __launch_bounds__(256) void sgc_spmm_kernel(
        const float* __restrict__ src,
        const int*   __restrict__ adj_row,
        const int*   __restrict__ adj_col,
        const float* __restrict__ adj_val,
        float* __restrict__ dst) {
    const int lane   = threadIdx.x & 31;
    const int gwave  = (blockIdx.x * blockDim.x + threadIdx.x) >> 5;
    const int nwaves = (gridDim.x * blockDim.x) >> 5;

    for (int e = gwave; e < N_EDGES; e += nwaves) {
        const int   r = adj_row[e];
        const int   c = adj_col[e];
        const float v = adj_val[e];
        const float2 s = *(const float2*)(src + (size_t)c * NCLASS + lane * 2);
        float* d = dst + (size_t)r * NCLASS + lane * 2;
        unsafeAtomicAdd(d,     s.x * v);
        unsafeAtomicAdd(d + 1, s.y * v);
    }
}

// ---------------------------------------------------------------------------
// Orchestration:  H0 = X@W^T ; H1 = A@H0 ; out = b + A@H1
// (A^2 X) W^T + b  ==  A^2 (X W^T) + b  -- projection commutes with A.
// ---------------------------------------------------------------------------
extern "C" void kernel_launch(void* const* d_in, const int* in_sizes, int n_in,
                              void* d_out, int out_size, void* d_ws, size_t ws_size,
                              hipStream_t stream) {
    const float* x    = (const float*)d_in[0];
    const int*   arow = (const int*)  d_in[1];
    const int*   acol = (const int*)  d_in[2];
    const float* aval = (const float*)d_in[3];
    const float* W    = (const float*)d_in[4];
    const float* bias = (const float*)d_in[5];
    float* out = (float*)d_out;

    float* h0 = (float*)d_ws;                       // [N_NODES, 64]
    float* h1 = h0 + (size_t)N_NODES * NCLASS;      // [N_NODES, 64]

    const int elems = N_NODES * NCLASS;
    sgc_init_kernel<<<(elems + 255) / 256, 256, 0, stream>>>(h1, out, bias);

    const int tiles  = N_NODES / 16;                // 6250, exact
    sgc_gemm_kernel<<<(tiles + 7) / 8, 256, 0, stream>>>(x, W, h0);

    sgc_spmm_kernel<<<2048, 256, 0, stream>>>(h0, arow, acol, aval, h1);
    sgc_spmm_kernel<<<2048, 256, 0, stream>>>(h1, arow, acol, aval, out);
}